// GINE_Net_62354335203921
// MI455X (gfx1250) — compile-verified
//
#include <hip/hip_runtime.h>
#include <hip/hip_bf16.h>

// ---------------------------------------------------------------------------
// GINE 2-layer GNN for MI455X (gfx1250, wave32, WMMA f32 16x16x4)
// ---------------------------------------------------------------------------

#define NNODES 50000
#define NEDGES 800000
#define INDIM  128
#define HID    64
#define BN_EPS 1e-5f

typedef __attribute__((ext_vector_type(2))) float v2f;
typedef __attribute__((ext_vector_type(8))) float v8f;

// D-matrix layout (16x16 f32, wave32):
//   lanes 0-15 : N = lane,     rows M = 0..7  in acc[0..7]
//   lanes 16-31: N = lane-16,  rows M = 8..15 in acc[0..7]
// A-matrix 16x4 f32: lane&15 = M, half-wave picks K {0,1} vs {2,3} (float2).
// B-matrix 4x16 f32: lane&15 = N, half-wave picks K {0,1} vs {2,3} (float2).

// ---------------------------------------------------------------------------
// Edge kernel: ea = edge_attr @ We^T + be ; m = relu(act[src] + ea) ;
//              atomic scatter-add m into agg[dst].   K (edge dim) = 64.
// NOUT = node feature dim (128 for layer 0, 64 for layer 1).
// One wave = 16 edges; 8 waves / block.
// ---------------------------------------------------------------------------
template<int NOUT>
__global__ __launch_bounds__(256)
void gine_edge_kernel(const float* __restrict__ act,        // [N, NOUT]
                      const float* __restrict__ edge_attr,  // [E, 64]
                      const float* __restrict__ We,         // [NOUT, 64] row-major
                      const float* __restrict__ be,         // [NOUT]
                      const int*   __restrict__ src,
                      const int*   __restrict__ dst,
                      float*       __restrict__ agg,        // [N, NOUT] pre-zeroed
                      int nEdges)
{
    const int lane  = threadIdx.x & 31;
    const int wave  = threadIdx.x >> 5;
    const int tile  = blockIdx.x * 8 + wave;
    const int ebase = tile * 16;
    if (ebase >= nEdges) return;              // wave-uniform: EXEC stays all-1s

    const int m     = lane & 15;
    const int ksel  = (lane >> 4) * 2;        // 0 or 2
    const int mbase = (lane >> 4) * 8;        // D rows owned by this lane
    const int nlane = lane & 15;

    // Preload all A fragments (K=64 -> 16 k-steps). edge_attr streams once and
    // exceeds L2: non-temporal so it doesn't evict the node table.
    v2f afrag[16];
    const v2f* arow = (const v2f*)(edge_attr + (size_t)(ebase + m) * 64 + ksel);
#pragma unroll
    for (int kk = 0; kk < 16; ++kk)
        afrag[kk] = __builtin_nontemporal_load(arow + kk * 2);

    int srcv[8], dstv[8];
#pragma unroll
    for (int i = 0; i < 8; ++i) {
        srcv[i] = src[ebase + mbase + i];
        dstv[i] = dst[ebase + mbase + i];
    }

#pragma unroll
    for (int nt = 0; nt < NOUT / 16; ++nt) {
        const int n0 = nt * 16;
        const float* wrow = We + (size_t)(n0 + nlane) * 64 + ksel;

        // Preload ALL B fragments before the WMMA chain: one load clause,
        // one wait, then 16 back-to-back v_wmma (no per-iteration stalls).
        v2f bfrag[16];
#pragma unroll
        for (int kk = 0; kk < 16; ++kk)
            bfrag[kk] = *(const v2f*)(wrow + kk * 4);

        v8f acc = {};
#pragma unroll
        for (int kk = 0; kk < 16; ++kk)
            acc = __builtin_amdgcn_wmma_f32_16x16x4_f32(
                      false, afrag[kk], false, bfrag[kk], (short)0, acc, false, false);

        const float bev = be[n0 + nlane];
#pragma unroll
        for (int i = 0; i < 8; ++i) {
            float xv = act[(size_t)srcv[i] * NOUT + n0 + nlane];
            float v  = acc[i] + bev + xv;
            v = v > 0.f ? v : 0.f;
            atomicAdd(agg + (size_t)dstv[i] * NOUT + n0 + nlane, v);
        }
    }
}

// ---------------------------------------------------------------------------
// Node MLP part 1:  t = (agg + x) @ W1^T + b1   (t: [N,64])
// Also accumulates per-feature sum and sum-of-squares for BatchNorm.
// K = node feature dim (128 layer 0, 64 layer 1).
// ---------------------------------------------------------------------------
template<int K>
__global__ __launch_bounds__(256)
void gine_node_mlp1(const float* __restrict__ agg,    // [N, K]
                    const float* __restrict__ x,      // [N, K]
                    const float* __restrict__ W1,     // [64, K]
                    const float* __restrict__ b1,     // [64]
                    float*       __restrict__ t,      // [N, 64]
                    float*       __restrict__ stats,  // [128] = sum[64],sumsq[64]
                    int nNodes)
{
    const int lane  = threadIdx.x & 31;
    const int wave  = threadIdx.x >> 5;
    const int tile  = blockIdx.x * 8 + wave;
    const int rbase = tile * 16;
    if (rbase >= nNodes) return;

    const int m     = lane & 15;
    const int ksel  = (lane >> 4) * 2;
    const int mbase = (lane >> 4) * 8;
    const int nlane = lane & 15;

    v2f afrag[K / 4];
    const float* ga = agg + (size_t)(rbase + m) * K + ksel;
    const float* xa = x   + (size_t)(rbase + m) * K + ksel;
#pragma unroll
    for (int kk = 0; kk < K / 4; ++kk) {
        v2f a = *(const v2f*)(ga + kk * 4);
        v2f b = *(const v2f*)(xa + kk * 4);
        afrag[kk] = a + b;
    }

#pragma unroll
    for (int nt = 0; nt < 4; ++nt) {            // 64 outputs = 4 n-tiles
        const int n0 = nt * 16;
        const float* wrow = W1 + (size_t)(n0 + nlane) * K + ksel;

        v2f bfrag[K / 4];
#pragma unroll
        for (int kk = 0; kk < K / 4; ++kk)
            bfrag[kk] = *(const v2f*)(wrow + kk * 4);

        v8f acc = {};
#pragma unroll
        for (int kk = 0; kk < K / 4; ++kk)
            acc = __builtin_amdgcn_wmma_f32_16x16x4_f32(
                      false, afrag[kk], false, bfrag[kk], (short)0, acc, false, false);

        const float bv = b1[n0 + nlane];
        float s = 0.f, s2 = 0.f;
#pragma unroll
        for (int i = 0; i < 8; ++i) {
            float v = acc[i] + bv;
            t[(size_t)(rbase + mbase + i) * 64 + n0 + nlane] = v;
            s  += v;
            s2 += v * v;
        }
        atomicAdd(stats +      n0 + nlane, s);
        atomicAdd(stats + 64 + n0 + nlane, s2);
    }
}

// ---------------------------------------------------------------------------
// Node MLP part 2:  out = relu(BN(t)) @ W2^T + b2  [+ optional trailing relu]
// BN scale/shift precomputed per block into LDS.
// ---------------------------------------------------------------------------
template<bool RELU_OUT>
__global__ __launch_bounds__(256)
void gine_node_mlp2(const float* __restrict__ t,      // [N, 64]
                    const float* __restrict__ stats,  // [128]
                    const float* __restrict__ g,      // [64]
                    const float* __restrict__ bb,     // [64]
                    const float* __restrict__ W2,     // [64, 64]
                    const float* __restrict__ b2,     // [64]
                    float*       __restrict__ out,    // [N, 64]
                    int nNodes)
{
    __shared__ float scaleL[64];
    __shared__ float shiftL[64];
    if (threadIdx.x < 64) {
        const int f  = threadIdx.x;
        const float invN = 1.0f / (float)nNodes;
        const float mu   = stats[f] * invN;
        const float var  = stats[64 + f] * invN - mu * mu;
        const float sc   = g[f] * rsqrtf(var + BN_EPS);
        scaleL[f] = sc;
        shiftL[f] = bb[f] - mu * sc;
    }
    __syncthreads();

    const int lane  = threadIdx.x & 31;
    const int wave  = threadIdx.x >> 5;
    const int tile  = blockIdx.x * 8 + wave;
    const int rbase = tile * 16;
    if (rbase >= nNodes) return;

    const int m     = lane & 15;
    const int ksel  = (lane >> 4) * 2;
    const int mbase = (lane >> 4) * 8;
    const int nlane = lane & 15;

    // A fragments = relu(t * scale + shift), built on the fly from LDS params
    v2f afrag[16];
    const float* ta = t + (size_t)(rbase + m) * 64 + ksel;
#pragma unroll
    for (int kk = 0; kk < 16; ++kk) {
        const int k = kk * 4 + ksel;
        v2f raw = *(const v2f*)(ta + kk * 4);
        float a0 = raw.x * scaleL[k]     + shiftL[k];
        float a1 = raw.y * scaleL[k + 1] + shiftL[k + 1];
        a0 = a0 > 0.f ? a0 : 0.f;
        a1 = a1 > 0.f ? a1 : 0.f;
        v2f a; a.x = a0; a.y = a1;
        afrag[kk] = a;
    }

#pragma unroll
    for (int nt = 0; nt < 4; ++nt) {
        const int n0 = nt * 16;
        const float* wrow = W2 + (size_t)(n0 + nlane) * 64 + ksel;

        v2f bfrag[16];
#pragma unroll
        for (int kk = 0; kk < 16; ++kk)
            bfrag[kk] = *(const v2f*)(wrow + kk * 4);

        v8f acc = {};
#pragma unroll
        for (int kk = 0; kk < 16; ++kk)
            acc = __builtin_amdgcn_wmma_f32_16x16x4_f32(
                      false, afrag[kk], false, bfrag[kk], (short)0, acc, false, false);

        const float bv = b2[n0 + nlane];
#pragma unroll
        for (int i = 0; i < 8; ++i) {
            float v = acc[i] + bv;
            if (RELU_OUT) v = v > 0.f ? v : 0.f;
            out[(size_t)(rbase + mbase + i) * 64 + n0 + nlane] = v;
        }
    }
}

// ---------------------------------------------------------------------------
// Launch
// ---------------------------------------------------------------------------
extern "C" void kernel_launch(void* const* d_in, const int* in_sizes, int n_in,
                              void* d_out, int out_size, void* d_ws, size_t ws_size,
                              hipStream_t stream)
{
    const float* x    = (const float*)d_in[0];
    const int*   ei   = (const int*)  d_in[1];   // [2, E] (int32; jax default x64 off)
    const float* ea   = (const float*)d_in[2];
    const float* We0  = (const float*)d_in[3];
    const float* be0  = (const float*)d_in[4];
    const float* W1_0 = (const float*)d_in[5];
    const float* b1_0 = (const float*)d_in[6];
    const float* g0   = (const float*)d_in[7];
    const float* bb0  = (const float*)d_in[8];
    const float* W2_0 = (const float*)d_in[9];
    const float* b2_0 = (const float*)d_in[10];
    const float* We1  = (const float*)d_in[11];
    const float* be1  = (const float*)d_in[12];
    const float* W1_1 = (const float*)d_in[13];
    const float* b1_1 = (const float*)d_in[14];
    const float* g1   = (const float*)d_in[15];
    const float* bb1  = (const float*)d_in[16];
    const float* W2_1 = (const float*)d_in[17];
    const float* b2_1 = (const float*)d_in[18];

    const int* src = ei;
    const int* dst = ei + NEDGES;

    // Workspace layout (floats)
    float* agg0   = (float*)d_ws;                        // N*128
    float* t0     = agg0 + (size_t)NNODES * INDIM;       // N*64
    float* h      = t0   + (size_t)NNODES * HID;         // N*64
    float* agg1   = h    + (size_t)NNODES * HID;         // N*64
    float* t1     = agg1 + (size_t)NNODES * HID;         // N*64
    float* stats0 = t1   + (size_t)NNODES * HID;         // 128
    float* stats1 = stats0 + 128;                        // 128

    // Zero atomic accumulators (every call; graph-capture safe)
    hipMemsetAsync(agg0, 0, (size_t)NNODES * INDIM * sizeof(float), stream);
    hipMemsetAsync(agg1, 0, (size_t)NNODES * HID * sizeof(float), stream);
    hipMemsetAsync(stats0, 0, 256 * sizeof(float), stream);

    const dim3 blk(256);
    const int edgeBlocks = NEDGES / (16 * 8);            // 6250 (exact)
    const int nodeBlocks = (NNODES / 16 + 7) / 8;        // 391 (3125 tiles)

    // ---- Layer 0 ----
    gine_edge_kernel<INDIM><<<edgeBlocks, blk, 0, stream>>>(
        x, ea, We0, be0, src, dst, agg0, NEDGES);
    gine_node_mlp1<INDIM><<<nodeBlocks, blk, 0, stream>>>(
        agg0, x, W1_0, b1_0, t0, stats0, NNODES);
    gine_node_mlp2<true><<<nodeBlocks, blk, 0, stream>>>(
        t0, stats0, g0, bb0, W2_0, b2_0, h, NNODES);     // trailing inter-layer relu

    // ---- Layer 1 ----
    gine_edge_kernel<HID><<<edgeBlocks, blk, 0, stream>>>(
        h, ea, We1, be1, src, dst, agg1, NEDGES);
    gine_node_mlp1<HID><<<nodeBlocks, blk, 0, stream>>>(
        agg1, h, W1_1, b1_1, t1, stats1, NNODES);
    gine_node_mlp2<false><<<nodeBlocks, blk, 0, stream>>>(
        t1, stats1, g1, bb1, W2_1, b2_1, (float*)d_out, NNODES);
}